// ChamferDistanceLoss_21028159881362
// MI455X (gfx1250) — compile-verified
//
#include <hip/hip_runtime.h>

typedef __attribute__((ext_vector_type(2))) float v2f;
typedef __attribute__((ext_vector_type(8))) float v8f;

#define KBINS 256
#define INF_BITS 0x7F800000u
#define TPW 8          // 16-target tiles per wave (128 targets/wave)

// ds_swizzle_b32 group-of-32 XOR pattern: offset = and_mask(0x1F) | xor<<10
#define SWZ_XOR(k) (0x1F | ((k) << 10))

template <int OFS>
__device__ __forceinline__ unsigned swz_u32(unsigned x) {
    return (unsigned)__builtin_amdgcn_ds_swizzle((int)x, OFS);
}
template <int OFS>
__device__ __forceinline__ float swz_f32(float x) {
    return __uint_as_float(
        (unsigned)__builtin_amdgcn_ds_swizzle((int)__float_as_uint(x), OFS));
}

__device__ __forceinline__ unsigned umin2(unsigned a, unsigned b) {
    return a < b ? a : b;   // v_min_u32
}

// One WMMA tile: D[i][j] = c_i - t_j (exact outer difference, no cancellation),
// then Q = D*D as an 8-wide vector fmul -> 4 v_pk_mul_f32 on the WMMA dest's
// adjacent pairs (no register gathering). q >= 0 so uint order == float order.
__device__ __forceinline__ void tile_sq(const v2f& Am, const v2f& Bm,
                                        unsigned q[8]) {
    v8f D = {};
    D = __builtin_amdgcn_wmma_f32_16x16x4_f32(
            false, Am, false, Bm, (short)0, D, false, false);
    v8f Q = D * D;
    #pragma unroll
    for (int r = 0; r < 8; ++r) q[r] = __float_as_uint(Q[r]);
}

__device__ __forceinline__ unsigned min8u(const unsigned q[8]) {
    return umin2(umin2(umin2(q[0], q[1]), umin2(q[2], q[3])),
                 umin2(umin2(q[4], q[5]), umin2(q[6], q[7])));
}

// ---------------------------------------------------------------------------
__global__ void chamfer_init_kernel(unsigned* __restrict__ gBins, int count) {
    int i = blockIdx.x * blockDim.x + threadIdx.x;
    if (i < count) gBins[i] = INF_BITS;
}

// ---------------------------------------------------------------------------
// One wave = TPW 16-target tiles vs all 256 centers.
//   A row i = [c_i, -1, 0, 0], B col j = [1, t_j, 0, 0] => D[i][j] = c_i - t_j
// ---------------------------------------------------------------------------
__global__ void chamfer_main_kernel(const float* __restrict__ target,
                                    const float* __restrict__ edges,
                                    unsigned* __restrict__ gBins,
                                    float* __restrict__ blockSums,
                                    int N) {
    __shared__ float    ldsC[KBINS];
    __shared__ unsigned ldsMin[KBINS];
    __shared__ float    waveSum[8];

    const int b   = blockIdx.y;
    const int tid = threadIdx.x;

    {   // 256 threads: bin centers from 257 edges + per-bin minima init
        float e0 = edges[b * (KBINS + 1) + tid];
        float e1 = edges[b * (KBINS + 1) + tid + 1];
        ldsC[tid]   = 0.5f * (e0 + e1);
        ldsMin[tid] = INF_BITS;
    }
    __syncthreads();

    const int lane = tid & 31;
    const int n    = lane & 15;   // column (target) index within tile
    const int hg   = lane >> 4;   // half-group (A/B K rows 0/1 vs 2/3)

    // scalar wave index -> all range tests are SGPR-uniform branches,
    // guaranteeing EXEC==all-ones at every WMMA.
    const int waveS = __builtin_amdgcn_readfirstlane(tid >> 5);
    const int base  = (blockIdx.x * 8 + waveS) * (TPW * 16);

    float localSum = 0.0f;
    if (base + TPW * 16 <= N) {
        // ---------------- hot path: fully branchless ----------------
        v2f Bm[TPW];
        #pragma unroll
        for (int tt = 0; tt < TPW; ++tt) {
            float tv = target[b * N + base + tt * 16 + n];
            Bm[tt].x = hg ? 0.0f : 1.0f;
            Bm[tt].y = hg ? 0.0f : tv;
        }

        unsigned tminU[TPW];
        #pragma unroll
        for (int tt = 0; tt < TPW; ++tt) tminU[tt] = INF_BITS;

        float cNext = ldsC[n];              // center for ct = 0
        #pragma unroll 1
        for (int ct = 0; ct < 16; ++ct) {   // 16 center tiles = 256 centers
            const float c = cNext;
            cNext = ldsC[(((ct + 1) & 15) * 16) + n];   // prefetch next ct
            v2f Am;
            Am.x = hg ? 0.0f : c;
            Am.y = hg ? 0.0f : -1.0f;

            unsigned q[8], rminU[8];
            // peel tile 0: rminU initialized directly (no INF mins)
            tile_sq(Am, Bm[0], q);
            #pragma unroll
            for (int r = 0; r < 8; ++r) rminU[r] = q[r];
            tminU[0] = umin2(tminU[0], min8u(q));

            #pragma unroll
            for (int tt = 1; tt < TPW; ++tt) {
                tile_sq(Am, Bm[tt], q);
                tminU[tt] = umin2(tminU[tt], min8u(q));
                #pragma unroll
                for (int r = 0; r < 8; ++r) rminU[r] = umin2(rminU[r], q[r]);
            }
            // direction 2: the LDS atomic unit IS the cross-lane combiner.
            // All 16 lanes of a half-group push their column-partial min at
            // the same address: 8 fire-and-forget ds_min_u32, no waits, no
            // swizzle tree, no exec masking.
            #pragma unroll
            for (int r = 0; r < 8; ++r)
                atomicMin(&ldsMin[ct * 16 + r + 8 * hg], rminU[r]);
        }

        // direction 1 finalize: level-wise xor16 (8 independent swizzles,
        // one wait), then sum tiles + serial lane-sum tree.
        unsigned t16[TPW];
        #pragma unroll
        for (int tt = 0; tt < TPW; ++tt)
            t16[tt] = swz_u32<SWZ_XOR(16)>(tminU[tt]);
        float s = 0.0f;
        #pragma unroll
        for (int tt = 0; tt < TPW; ++tt)
            s += __uint_as_float(umin2(tminU[tt], t16[tt]));
        s += swz_f32<SWZ_XOR(1)>(s);
        s += swz_f32<SWZ_XOR(2)>(s);
        s += swz_f32<SWZ_XOR(4)>(s);
        s += swz_f32<SWZ_XOR(8)>(s);   // both 16-lane groups hold equal sums
        localSum = (lane == 0) ? s : 0.0f;
    } else if (base < N) {
        // ------------- cold generic tail (no WMMA, rarely/never runs) -------
        const int jEnd = min(base + TPW * 16, N);
        float s = 0.0f;
        for (int j = base + lane; j < jEnd; j += 32) {
            float tv = target[b * N + j];
            float tmin = __uint_as_float(INF_BITS);
            for (int k = 0; k < KBINS; ++k) {
                float d  = ldsC[k] - tv;
                float d2 = d * d;
                tmin = fminf(tmin, d2);
                atomicMin(&ldsMin[k], __float_as_uint(d2));
            }
            s += tmin;
        }
        s += swz_f32<SWZ_XOR(1)>(s);
        s += swz_f32<SWZ_XOR(2)>(s);
        s += swz_f32<SWZ_XOR(4)>(s);
        s += swz_f32<SWZ_XOR(8)>(s);
        s += swz_f32<SWZ_XOR(16)>(s);  // sum over all 32 lanes
        localSum = (lane == 0) ? s : 0.0f;
    }

    if (lane == 0) waveSum[waveS] = localSum;
    __syncthreads();

    if (tid == 0) {
        float s = 0.0f;
        #pragma unroll
        for (int w = 0; w < 8; ++w) s += waveSum[w];
        blockSums[b * gridDim.x + blockIdx.x] = s;   // fixed slot: deterministic
    }

    // export per-bin minima (idempotent uint min, order-independent)
    atomicMin(&gBins[b * KBINS + tid], ldsMin[tid]);
}

// ---------------------------------------------------------------------------
__global__ void chamfer_final_kernel(const unsigned* __restrict__ gBins,
                                     const float* __restrict__ blockSums,
                                     int nBatches, int nBlockSums,
                                     float* __restrict__ out, float invB) {
    __shared__ float red[256];
    const int t = threadIdx.x;
    float s = 0.0f;
    for (int bb = 0; bb < nBatches; ++bb)
        s += __uint_as_float(gBins[bb * KBINS + t]);
    for (int i = t; i < nBlockSums; i += 256)
        s += blockSums[i];
    red[t] = s;
    __syncthreads();
    for (int w = 128; w > 0; w >>= 1) {
        if (t < w) red[t] += red[t + w];
        __syncthreads();
    }
    if (t == 0) out[0] = red[0] * invB;
}

// ---------------------------------------------------------------------------
extern "C" void kernel_launch(void* const* d_in, const int* in_sizes, int n_in,
                              void* d_out, int out_size, void* d_ws, size_t ws_size,
                              hipStream_t stream) {
    const float* target = (const float*)d_in[0];   // [B,1,240,320] f32
    const float* edges  = (const float*)d_in[1];   // [B,257] f32

    const int B = in_sizes[1] / (KBINS + 1);       // 2
    const int N = in_sizes[0] / B;                 // 76800
    const int jobs    = (N + TPW * 16 - 1) / (TPW * 16);   // 128-target wave jobs
    const int blocksX = (jobs + 7) / 8;                    // 75

    unsigned* gBins     = (unsigned*)d_ws;                 // B*256 uints
    float*    blockSums = (float*)((char*)d_ws + 2048);    // B*blocksX floats
    float*    out       = (float*)d_out;

    chamfer_init_kernel<<<(B * KBINS + 255) / 256, 256, 0, stream>>>(gBins, B * KBINS);

    dim3 grid(blocksX, B);
    chamfer_main_kernel<<<grid, 256, 0, stream>>>(target, edges, gBins, blockSums, N);

    chamfer_final_kernel<<<1, 256, 0, stream>>>(gBins, blockSums, B, B * blocksX,
                                                out, 1.0f / (float)B);
}